// PtychoAD_vec_7017976561689
// MI455X (gfx1250) — compile-verified
//
#include <hip/hip_runtime.h>

typedef float v2f __attribute__((ext_vector_type(2)));
typedef float v8f __attribute__((ext_vector_type(8)));

#define PI2F 6.28318530717958647692f

static const int NYX   = 256;
static const int IMG   = 256 * 256;   // 65536
static const int BATCH = 64;
static const int PMODE = 4;
static const int NZ    = 8;
static const int CB    = 8;           // batches per chunk
static const int CI    = CB * PMODE;  // 32 images per chunk
static const int NCHUNK = BATCH / CB; // 8

// async global->LDS 16B copy (CDNA5; tracked by ASYNCcnt)
#define ASYNC_B128(ldsaddr, gaddr)                                             \
    asm volatile("global_load_async_to_lds_b128 %0, %1, off"                   \
                 :: "v"((unsigned)(unsigned long long)(ldsaddr)),              \
                    "v"((unsigned long long)(gaddr)) : "memory")

#define WAIT_ASYNC0() asm volatile("s_wait_asynccnt 0x0" ::: "memory")

// ---------------------------------------------------------------------------
// Build symmetric DFT matrices: Wf = exp(-2*pi*i*k*n/256), Wi = exp(+...)/256
// ---------------------------------------------------------------------------
__global__ void k_init_w(float* __restrict__ WfR, float* __restrict__ WfI,
                         float* __restrict__ WiR, float* __restrict__ WiI) {
    int idx = blockIdx.x * blockDim.x + threadIdx.x;   // 65536 threads
    int k = idx >> 8, n = idx & 255;
    int t = (k * n) & 255;                              // exact phase wrap
    float ang = -PI2F * (float)t * (1.0f / 256.0f);
    float s, c;
    __sincosf(ang, &s, &c);
    WfR[idx] = c;
    WfI[idx] = s;
    WiR[idx] = c * (1.0f / 256.0f);
    WiI[idx] = -s * (1.0f / 256.0f);
}

// interleaved complex -> planar
__global__ void k_split(const float* __restrict__ src,
                        float* __restrict__ dstR, float* __restrict__ dstI, int n) {
    int i = blockIdx.x * blockDim.x + threadIdx.x;
    if (i < n) { dstR[i] = src[2 * i]; dstI[i] = src[2 * i + 1]; }
}

// ---------------------------------------------------------------------------
// Complex 256x256 matmul per image, LDS-staged + double-buffered async copy.
//   O[img] = X[img?] x Y[img?]    (stride 0 => broadcast operand, e.g. DFT W)
// Block: 512 threads = 16 waves; block computes a 64x64 output block;
// each wave owns one 16x16 tile via V_WMMA_F32_16X16X4_F32 (4 chains, K=256).
// K is processed in 8 chunks of 32; panels staged to LDS with
// GLOBAL_LOAD_ASYNC_TO_LDS_B128 and consumed via ds_load.
// ---------------------------------------------------------------------------
#define XPITCH 36   // 64x32 panel rows padded: 144 B rows (16B aligned, banks spread)
#define YPITCH 80   // 32x64 panel rows padded: 320 B rows (16B aligned, halves split)

__global__ __launch_bounds__(512) void k_cmatmul(
    const float* __restrict__ XR, const float* __restrict__ XI, int xs,
    const float* __restrict__ YR, const float* __restrict__ YI, int ys,
    float* __restrict__ outR, float* __restrict__ outI, int os)
{
    __shared__ float XsR[2][64][XPITCH];
    __shared__ float XsI[2][64][XPITCH];
    __shared__ float YsR[2][32][YPITCH];
    __shared__ float YsI[2][32][YPITCH];

    const int tid  = threadIdx.x;
    const int lane = tid & 31;
    const int w    = tid >> 5;                 // wave 0..15
    const int img  = blockIdx.x >> 4;          // 16 blocks per image
    const int blk  = blockIdx.x & 15;
    const int mblk = (blk >> 2) << 6;          // 64-row block base
    const int nblk = (blk & 3) << 6;           // 64-col block base

    const float* xr = XR + (size_t)img * (size_t)xs;
    const float* xi = XI + (size_t)img * (size_t)xs;
    const float* yr = YR + (size_t)img * (size_t)ys;
    const float* yi = YI + (size_t)img * (size_t)ys;

    // per-wave tile inside the 64x64 block
    const int mrow0 = (w >> 2) << 4;           // local m-tile row base
    const int ntl   = (w & 3) << 4;            // local n-tile col base
    const int m     = lane & 15;
    const int kk    = (lane >> 4) << 1;        // K pair select {0,1} vs {2,3}

    // staging assignment: one b128 per thread per plane per chunk
    const int xrow = tid >> 3, xc4 = (tid & 7) << 2;    // 64 x 8 slots
    const int yrow = tid >> 4, yc4 = (tid & 15) << 2;   // 32 x 16 slots

    auto stage = [&](int buf, int kc) {
        ASYNC_B128(&XsR[buf][xrow][xc4], xr + (mblk + xrow) * 256 + kc + xc4);
        ASYNC_B128(&XsI[buf][xrow][xc4], xi + (mblk + xrow) * 256 + kc + xc4);
        ASYNC_B128(&YsR[buf][yrow][yc4], yr + (kc + yrow) * 256 + nblk + yc4);
        ASYNC_B128(&YsI[buf][yrow][yc4], yi + (kc + yrow) * 256 + nblk + yc4);
    };

    v8f cRR = {}; v8f cII = {}; v8f cRI = {}; v8f cIR = {};

    stage(0, 0);
    WAIT_ASYNC0();
    __syncthreads();

    for (int c = 0; c < 8; ++c) {
        const int buf = c & 1;
        if (c < 7) stage(buf ^ 1, (c + 1) << 5);

        #pragma unroll
        for (int k0 = 0; k0 < 32; k0 += 4) {
            // A fragment 16x4 from X panel (ds_load_b64)
            v2f aR = *(const v2f*)&XsR[buf][mrow0 + m][k0 + kk];
            v2f aI = *(const v2f*)&XsI[buf][mrow0 + m][k0 + kk];
            // B fragment 4x16 from Y panel (ds_load_b32)
            v2f bR, bI;
            bR.x = YsR[buf][k0 + kk + 0][ntl + m];
            bR.y = YsR[buf][k0 + kk + 1][ntl + m];
            bI.x = YsI[buf][k0 + kk + 0][ntl + m];
            bI.y = YsI[buf][k0 + kk + 1][ntl + m];

            cRR = __builtin_amdgcn_wmma_f32_16x16x4_f32(false, aR, false, bR, (short)0, cRR, false, false);
            cII = __builtin_amdgcn_wmma_f32_16x16x4_f32(false, aI, false, bI, (short)0, cII, false, false);
            cRI = __builtin_amdgcn_wmma_f32_16x16x4_f32(false, aR, false, bI, (short)0, cRI, false, false);
            cIR = __builtin_amdgcn_wmma_f32_16x16x4_f32(false, aI, false, bR, (short)0, cIR, false, false);
        }

        WAIT_ASYNC0();       // next buffer's panels have landed
        __syncthreads();     // everyone done reading buf before it is reused
    }

    v8f cr = cRR - cII;      // F32 WMMA lacks A/B negate -> VALU combine
    v8f ci = cRI + cIR;

    float* orp = outR + (size_t)img * (size_t)os;
    float* oip = outI + (size_t)img * (size_t)os;
    const int mt = mblk + mrow0;
    const int nt = nblk + ntl;
    const int rbase = mt + ((lane >> 4) << 3);
    #pragma unroll
    for (int v = 0; v < 8; ++v) {
        int o = (rbase + v) * 256 + nt + m;
        orp[o] = cr[v];
        oip[o] = ci[v];
    }
}

// ---------------------------------------------------------------------------
// spec = fft2(probe)[p] * exp(-2*pi*i*(sy*y + sx*x)/256)  for chunk images
// ---------------------------------------------------------------------------
__global__ void k_build_spec(const float* __restrict__ FpR, const float* __restrict__ FpI,
                             const float* __restrict__ shifts, const int* __restrict__ indices,
                             int chunk, float* __restrict__ outR, float* __restrict__ outI) {
    int gid = blockIdx.x * blockDim.x + threadIdx.x;   // CI*IMG threads
    int i = gid >> 16;                                 // image in chunk
    int e = gid & 65535;
    int y = e >> 8, x = e & 255;
    int lb = i >> 2, p = i & 3;
    int b  = chunk * CB + lb;
    int pi = indices[b];
    float sy = shifts[2 * pi], sx = shifts[2 * pi + 1];
    float ph = -PI2F * (sy * (float)y + sx * (float)x) * (1.0f / 256.0f);
    float s, c;
    __sincosf(ph, &s, &c);
    float fr = FpR[p * IMG + e], fi = FpI[p * IMG + e];
    outR[gid] = fr * c - fi * s;
    outI[gid] = fr * s + fi * c;
}

// gather object patch for slice z: obj = amp * exp(i*pha)
__global__ void k_obj_slice(const float* __restrict__ obj_ap,
                            const int* __restrict__ crop, const int* __restrict__ indices,
                            int chunk, int z,
                            float* __restrict__ objR, float* __restrict__ objI) {
    int gid = blockIdx.x * blockDim.x + threadIdx.x;   // CB*IMG threads
    int lb = gid >> 16;
    int e  = gid & 65535;
    int y = e >> 8, x = e & 255;
    int b  = chunk * CB + lb;
    int pi = indices[b];
    int h = crop[2 * pi] + y;
    int w = crop[2 * pi + 1] + x;
    size_t o = (((size_t)z * 1024 + (size_t)h) * 1024 + (size_t)w) * 2;  // OMODE==1
    float amp = obj_ap[o], pha = obj_ap[o + 1];
    float s, c;
    __sincosf(pha, &s, &c);
    objR[gid] = amp * c;
    objI[gid] = amp * s;
}

__global__ void k_obj_mult(float* __restrict__ pR, float* __restrict__ pI,
                           const float* __restrict__ oR, const float* __restrict__ oI) {
    int gid = blockIdx.x * blockDim.x + threadIdx.x;   // CI*IMG threads
    int i = gid >> 16;
    int e = gid & 65535;
    int lb = i >> 2;
    float orr = oR[lb * IMG + e], oii = oI[lb * IMG + e];
    float a = pR[gid], bb = pI[gid];
    pR[gid] = a * orr - bb * oii;
    pI[gid] = a * oii + bb * orr;
}

__global__ void k_hmult(float* __restrict__ xR, float* __restrict__ xI,
                        const float* __restrict__ H) {
    int gid = blockIdx.x * blockDim.x + threadIdx.x;   // CI*IMG threads
    int e = gid & 65535;
    float hr = H[2 * e], hi = H[2 * e + 1];
    float a = xR[gid], bb = xI[gid];
    xR[gid] = a * hr - bb * hi;
    xI[gid] = a * hi + bb * hr;
}

// dp[b][fftshift(y,x)] = sum_p |X|^2
__global__ void k_power(const float* __restrict__ xR, const float* __restrict__ xI,
                        float* __restrict__ dp, int chunk) {
    int gid = blockIdx.x * blockDim.x + threadIdx.x;   // CB*IMG threads
    int lb = gid >> 16;
    int e  = gid & 65535;
    int y = e >> 8, x = e & 255;
    int b = chunk * CB + lb;
    float acc = 0.0f;
    #pragma unroll
    for (int p = 0; p < PMODE; ++p) {
        int i = (lb * PMODE + p) * IMG + e;
        float r = xR[i], im = xI[i];
        acc += r * r + im * im;
    }
    int ys = (y + 128) & 255, xs = (x + 128) & 255;
    dp[((size_t)b << 16) + (ys << 8) + xs] = acc;
}

// ---------------------------------------------------------------------------
extern "C" void kernel_launch(void* const* d_in, const int* in_sizes, int n_in,
                              void* d_out, int out_size, void* d_ws, size_t ws_size,
                              hipStream_t stream) {
    (void)in_sizes; (void)n_in; (void)out_size; (void)ws_size;

    const float* obj_ap  = (const float*)d_in[0];
    const float* probe   = (const float*)d_in[1];   // interleaved complex64
    const float* shifts  = (const float*)d_in[2];
    const float* H       = (const float*)d_in[3];   // interleaved complex64
    const int*   crop    = (const int*)d_in[4];
    const int*   indices = (const int*)d_in[5];
    float* dp = (float*)d_out;

    float* w = (float*)d_ws;
    float* WfR = w; w += IMG;
    float* WfI = w; w += IMG;
    float* WiR = w; w += IMG;
    float* WiI = w; w += IMG;
    float* FpR = w; w += PMODE * IMG;
    float* FpI = w; w += PMODE * IMG;
    float* AR  = w; w += (size_t)CI * IMG;
    float* AI  = w; w += (size_t)CI * IMG;
    float* BR  = w; w += (size_t)CI * IMG;
    float* BI  = w; w += (size_t)CI * IMG;
    float* PR  = w; w += (size_t)CI * IMG;
    float* PIm = w; w += (size_t)CI * IMG;
    float* OJR = w; w += (size_t)CB * IMG;
    float* OJI = w; w += (size_t)CB * IMG;

    const dim3 blk(256);
    const dim3 mmblk(512);

    // DFT matrices + fft2(probe) once (deterministic every call)
    k_init_w<<<IMG / 256, blk, 0, stream>>>(WfR, WfI, WiR, WiI);
    k_split<<<(PMODE * IMG) / 256, blk, 0, stream>>>(probe, AR, AI, PMODE * IMG);
    k_cmatmul<<<PMODE * 16, mmblk, 0, stream>>>(WfR, WfI, 0, AR, AI, IMG, BR, BI, IMG);
    k_cmatmul<<<PMODE * 16, mmblk, 0, stream>>>(BR, BI, IMG, WfR, WfI, 0, FpR, FpI, IMG);

    for (int c = 0; c < NCHUNK; ++c) {
        // psi0 = ifft2( fft2(probe) * ramp )
        k_build_spec<<<(CI * IMG) / 256, blk, 0, stream>>>(FpR, FpI, shifts, indices, c, AR, AI);
        k_cmatmul<<<CI * 16, mmblk, 0, stream>>>(WiR, WiI, 0, AR, AI, IMG, BR, BI, IMG);
        k_cmatmul<<<CI * 16, mmblk, 0, stream>>>(BR, BI, IMG, WiR, WiI, 0, PR, PIm, IMG);

        for (int z = 0; z < NZ; ++z) {
            k_obj_slice<<<(CB * IMG) / 256, blk, 0, stream>>>(obj_ap, crop, indices, c, z, OJR, OJI);
            k_obj_mult<<<(CI * IMG) / 256, blk, 0, stream>>>(PR, PIm, OJR, OJI);
            // psi = ifft2( fft2(psi) * H )
            k_cmatmul<<<CI * 16, mmblk, 0, stream>>>(WfR, WfI, 0, PR, PIm, IMG, AR, AI, IMG);
            k_cmatmul<<<CI * 16, mmblk, 0, stream>>>(AR, AI, IMG, WfR, WfI, 0, BR, BI, IMG);
            k_hmult<<<(CI * IMG) / 256, blk, 0, stream>>>(BR, BI, H);
            k_cmatmul<<<CI * 16, mmblk, 0, stream>>>(WiR, WiI, 0, BR, BI, IMG, AR, AI, IMG);
            k_cmatmul<<<CI * 16, mmblk, 0, stream>>>(AR, AI, IMG, WiR, WiI, 0, PR, PIm, IMG);
        }

        // dp = sum_p |fftshift(fft2(psi))|^2
        k_cmatmul<<<CI * 16, mmblk, 0, stream>>>(WfR, WfI, 0, PR, PIm, IMG, AR, AI, IMG);
        k_cmatmul<<<CI * 16, mmblk, 0, stream>>>(AR, AI, IMG, WfR, WfI, 0, BR, BI, IMG);
        k_power<<<(CB * IMG) / 256, blk, 0, stream>>>(BR, BI, dp, c);
    }
}